// CrossScalePeriodicFeatureAggregator_40810779247485
// MI455X (gfx1250) — compile-verified
//
#include <hip/hip_runtime.h>
#include <stdint.h>

// Problem constants (from reference): E=8 experts, B=128, L=96, D=512
#define NE 8
#define NB 128
#define NL 96
#define ND 512
#define LD (NL * ND)                 // 49152 floats per (b)
#define CHUNK 2048                   // floats per expert row per tile (8 KB)
#define CHUNKS_PER_B (LD / CHUNK)    // 24
#define SLOTS (CHUNK / 4)            // 512 float4 slots per expert row
#define EXPERT_STRIDE (NB * LD)      // 6291456 elements between expert planes

typedef unsigned int u32;
typedef __attribute__((ext_vector_type(4))) float v4f;
typedef __attribute__((ext_vector_type(4))) u32   v4u;
typedef __attribute__((ext_vector_type(8))) int   v8i;
typedef __attribute__((ext_vector_type(4))) int   v4i;

#if defined(__HIP_DEVICE_COMPILE__) && __has_builtin(__builtin_amdgcn_tensor_load_to_lds)
#define USE_TDM 1
#else
#define USE_TDM 0
#endif

#if USE_TDM
// Issue one Tensor Data Mover DMA: 2D tile (tile_dim0=CHUNK floats along d,
// tile_dim1=NE expert rows, row stride = EXPERT_STRIDE elements) -> LDS,
// packed contiguously, one CHUNK-float row per expert. Descriptor per CDNA5 ISA D# layout.
// (Round-1 disasm confirmed the encoding: s{1,0,addr_lo,addr_hi|type} /
//  s{0x20000, 0x08000000, 0x80000, 0x08000000, 8, 0x600000, 0, 0}.)
__device__ __forceinline__ void tdm_load_tile(uint64_t gaddr, u32 lds_byte_off) {
  // ---- D# group 0 (128b): flags / lds_addr / global_addr / type ----
  v4u g0;
  g0.x = 1u;                                   // count=1, user mode, no gather
  g0.y = lds_byte_off;                         // lds_addr (bytes)
  g0.z = (u32)(gaddr & 0xFFFFFFFFull);         // global_addr[31:0]
  g0.w = (u32)((gaddr >> 32) & 0x01FFFFFFull)  // global_addr[56:32]
       | (2u << 30);                           // type = 2 ("image")

  // ---- D# group 1 (256b): sizes / strides ----
  const u32 tdim0   = CHUNK;          // tensor_dim0 (elements)
  const u32 tdim1   = NE;             // tensor_dim1
  const u32 tile0   = CHUNK;          // tile_dim0
  const u32 tile1   = NE;             // tile_dim1 (tile_dim2 = 0 -> 2D)
  const u32 stride0 = EXPERT_STRIDE;  // tensor_dim0_stride (elements)

  v8i g1;
  g1[0] = (int)(2u << 16);                                  // data_size=2 -> 4B
  g1[1] = (int)((tdim0 & 0xFFFFu) << 16);                   // tensor_dim0[15:0]
  g1[2] = (int)((tdim0 >> 16) | ((tdim1 & 0xFFFFu) << 16)); // dim0 hi | dim1 lo
  g1[3] = (int)((tdim1 >> 16) | (tile0 << 16));             // dim1 hi | tile_dim0
  g1[4] = (int)(tile1);                                     // tile_dim1 | tile_dim2=0
  g1[5] = (int)(stride0);                                   // dim0_stride[31:0]
  g1[6] = 0;                                                // dim0_stride hi | dim1_stride lo
  g1[7] = 0;                                                // dim1_stride hi

  v4i gz = {0, 0, 0, 0};   // groups 2/3 unused (2D tile)
#if __has_include(<hip/amd_detail/amd_gfx1250_TDM.h>)
  // amdgpu-toolchain (clang-23 / therock): 6-arg builtin
  v8i gz8 = {0, 0, 0, 0, 0, 0, 0, 0};
  __builtin_amdgcn_tensor_load_to_lds(g0, g1, gz, gz, gz8, 0);
#else
  // ROCm 7.2 (clang-22): 5-arg builtin
  __builtin_amdgcn_tensor_load_to_lds(g0, g1, gz, gz, 0);
#endif
}
#endif // USE_TDM

__global__ __launch_bounds__(256) void
moe_exp_combine_log_kernel(const float* __restrict__ xs,
                           const float* __restrict__ gates,
                           float* __restrict__ out) {
  __shared__ float tile[NE * CHUNK]; // 64 KB -> 5 blocks/WGP (320 KB LDS)

  const u32 blk = blockIdx.x;
  const u32 b   = blk / CHUNKS_PER_B;   // batch index (wave-uniform)
  const u32 c   = blk % CHUNKS_PER_B;   // chunk within (l,d) plane

  const uint64_t tile_elem = (uint64_t)b * LD + (uint64_t)c * CHUNK;

#if USE_TDM
  // One TDM DMA per block gathers the 8 strided expert rows into LDS.
  if (threadIdx.x < 32) {
    tdm_load_tile((uint64_t)(uintptr_t)xs + 4ull * tile_elem, 0u);
  }
  __builtin_amdgcn_s_wait_tensorcnt(0); // issuing wave drains TENSORcnt; others pass
  __syncthreads();                      // publish LDS tile to all 8 waves

  // The TDM engine wrote `tile` behind the compiler's back: escape the LDS
  // address into an opaque asm with a memory clobber so clang cannot fold the
  // subsequent ds_loads of a "never-stored" shared array into undef/NaN
  // (which is exactly what round 1 did: ds=0, v_s_log_f32 of 0x7fc00000).
  {
    const float* esc = &tile[0];
    asm volatile("" : "+v"(esc) : : "memory");
  }
#else
  // Fallback: cooperative vector loads into LDS (same layout).
  {
    const v4f* src = (const v4f*)(xs + tile_elem);
    v4f* dst = (v4f*)tile;
    for (u32 i = threadIdx.x; i < NE * SLOTS; i += 256) {
      const u32 e = i / SLOTS, s = i % SLOTS;
      dst[i] = src[(uint64_t)e * (EXPERT_STRIDE / 4) + s];
    }
    __syncthreads();
  }
#endif

  // Gate row for this batch (wave-uniform -> scalar loads)
  float g[NE];
#pragma unroll
  for (int e = 0; e < NE; ++e) g[e] = gates[b * NE + e];

  const v4f* lds4 = (const v4f*)tile;
  v4f* out4 = (v4f*)out;
  const u32 out_base = b * (LD / 4) + c * SLOTS;

  const float eps = 2.2204460492503131e-16f; // float(np.finfo(np.float64).eps)

#pragma unroll
  for (int rep = 0; rep < SLOTS / 256; ++rep) {
    const u32 s = threadIdx.x + rep * 256u;

    float a0 = 0.f, a1 = 0.f, a2 = 0.f, a3 = 0.f;
#pragma unroll
    for (int e = 0; e < NE; ++e) {
      const v4f v = lds4[e * SLOTS + s];   // ds_load_b128, conflict-free
      a0 = fmaf(g[e], __expf(v.x), a0);    // v_exp_f32 + v_fma_f32
      a1 = fmaf(g[e], __expf(v.y), a1);
      a2 = fmaf(g[e], __expf(v.z), a2);
      a3 = fmaf(g[e], __expf(v.w), a3);
    }

    v4f r;
    r.x = __logf(a0 == 0.f ? eps : a0);
    r.y = __logf(a1 == 0.f ? eps : a1);
    r.z = __logf(a2 == 0.f ? eps : a2);
    r.w = __logf(a3 == 0.f ? eps : a3);

    // Output is write-once/never re-read: non-temporal b128 store keeps it
    // out of L2 so the streamed xs tensor owns the cache.
    __builtin_nontemporal_store(r, &out4[out_base + s]);
  }
}

extern "C" void kernel_launch(void* const* d_in, const int* in_sizes, int n_in,
                              void* d_out, int out_size, void* d_ws, size_t ws_size,
                              hipStream_t stream) {
  (void)in_sizes; (void)n_in; (void)out_size; (void)d_ws; (void)ws_size;
  const float* xs    = (const float*)d_in[0];  // [E,B,L,D] fp32
  const float* gates = (const float*)d_in[1];  // [B,E]     fp32
  float*       out   = (float*)d_out;          // [B,L,D]   fp32

  const int grid = NB * CHUNKS_PER_B;          // 3072 blocks, 1 tile each
  moe_exp_combine_log_kernel<<<grid, 256, 0, stream>>>(xs, gates, out);
}